// Model_68152541052992
// MI455X (gfx1250) — compile-verified
//
#include <hip/hip_runtime.h>
#include <hip/hip_bf16.h>

typedef __attribute__((ext_vector_type(2))) float v2f;
typedef __attribute__((ext_vector_type(8))) float v8f;

#define E_ 32
#define M_ 1024
#define K_ 512
#define N_ 512

#define BM 128   // block tile M  (2 waves)
#define BN 256   // block tile N  (4 waves)
#define WM 64    // wave tile M   (4 x 16)
#define WN 64    // wave tile N   (4 x 16)

// C[e] = A[e] (MxK) * B[e]^T (B is NxK), rows m >= cnt[e] forced to zero.
//
// fp32 WMMA fragment mapping (V_WMMA_F32_16X16X4_F32):
//   A (16x4):  lane lo holds row m=lo; vgpr pair = {K=2*hi, K=2*hi+1}  -> float2 along K
//   B (4x16):  lane lo holds col n=lo; vgpr pair = {K=2*hi, K=2*hi+1}  -> float2 along K
//   (B is consumed K-major; since C = A*B^T, stored row-major B[n][k] is already K-major)
//   C/D (16x16): lane lo = col n, vgpr r = row (r + 8*hi)
__global__ __launch_bounds__(256) void expert_gemm_f32_wmma(
    const float* __restrict__ A,
    const float* __restrict__ B,
    const int*   __restrict__ cnts,
    float*       __restrict__ C)
{
    const int e    = blockIdx.z;
    const int lane = threadIdx.x & 31;
    const int wave = threadIdx.x >> 5;
    const int wm   = wave & 1;   // 0..1
    const int wn   = wave >> 1;  // 0..3

    const int m_w = blockIdx.y * BM + wm * WM;
    const int n_w = blockIdx.x * BN + wn * WN;

    const int lo = lane & 15;
    const int hi = lane >> 4;

    const int cnt = cnts[e];

    const float* __restrict__ Ae = A + (size_t)e * M_ * K_;
    const float* __restrict__ Be = B + (size_t)e * N_ * K_;

    // Per-fragment row base pointers (each lane reads a contiguous float2 at +k0)
    const float* aRow[4];
    const float* bRow[4];
#pragma unroll
    for (int mi = 0; mi < 4; ++mi)
        aRow[mi] = Ae + (size_t)(m_w + 16 * mi + lo) * K_ + 2 * hi;
#pragma unroll
    for (int ni = 0; ni < 4; ++ni)
        bRow[ni] = Be + (size_t)(n_w + 16 * ni + lo) * K_ + 2 * hi;

    v8f acc[4][4];
#pragma unroll
    for (int mi = 0; mi < 4; ++mi)
#pragma unroll
        for (int ni = 0; ni < 4; ++ni)
            acc[mi][ni] = (v8f)(0.0f);

    // Skip the whole K loop if every row of this wave's 64-row tile is masked.
    if (m_w < cnt) {
#pragma unroll 2
        for (int k0 = 0; k0 < K_; k0 += 4) {
            v2f af[4], bf[4];
#pragma unroll
            for (int mi = 0; mi < 4; ++mi)
                af[mi] = *(const v2f*)(aRow[mi] + k0);
#pragma unroll
            for (int ni = 0; ni < 4; ++ni)
                bf[ni] = *(const v2f*)(bRow[ni] + k0);

#pragma unroll
            for (int mi = 0; mi < 4; ++mi)
#pragma unroll
                for (int ni = 0; ni < 4; ++ni)
                    acc[mi][ni] = __builtin_amdgcn_wmma_f32_16x16x4_f32(
                        /*neg_a=*/false, af[mi],
                        /*neg_b=*/false, bf[ni],
                        /*c_mod=*/(short)0, acc[mi][ni],
                        /*reuse_a=*/false, /*reuse_b=*/false);
        }
    }

    // Store with per-row validity mask (must write zeros: d_out is poisoned).
    float* __restrict__ Ce = C + (size_t)e * M_ * N_;
#pragma unroll
    for (int mi = 0; mi < 4; ++mi) {
#pragma unroll
        for (int ni = 0; ni < 4; ++ni) {
            const int n = n_w + 16 * ni + lo;
#pragma unroll
            for (int r = 0; r < 8; ++r) {
                const int m = m_w + 16 * mi + 8 * hi + r;
                const float v = (m < cnt) ? acc[mi][ni][r] : 0.0f;
                Ce[(size_t)m * N_ + n] = v;
            }
        }
    }
}

extern "C" void kernel_launch(void* const* d_in, const int* in_sizes, int n_in,
                              void* d_out, int out_size, void* d_ws, size_t ws_size,
                              hipStream_t stream) {
    const float* A    = (const float*)d_in[0];
    const float* B    = (const float*)d_in[1];
    const int*   cnts = (const int*)d_in[2];
    float*       C    = (float*)d_out;

    dim3 grid(N_ / BN, M_ / BM, E_);  // (2, 8, 32)
    dim3 block(256, 1, 1);            // 8 waves (wave32)
    expert_gemm_f32_wmma<<<grid, block, 0, stream>>>(A, B, cnts, C);
}